// GINEncoder_54571854463790
// MI455X (gfx1250) — compile-verified
//
#include <hip/hip_runtime.h>

// GIN encoder, 2 layers, HID = IN_DIM = 128, fp32 end-to-end.
// WMMA: V_WMMA_F32_16X16X4_F32 (full fp32 precision, matches reference).

#define HIDF 128

typedef __attribute__((ext_vector_type(2))) float v2f;
typedef __attribute__((ext_vector_type(8))) float v8f;

// ---------------------------------------------------------------- elementwise
__global__ __launch_bounds__(256) void copy_f32(const float* __restrict__ in,
                                                float* __restrict__ out, int n) {
    int i = blockIdx.x * 256 + threadIdx.x;
    if (i < n) out[i] = in[i];
}

__global__ __launch_bounds__(256) void zero_f32(float* __restrict__ p, int n) {
    int i = blockIdx.x * 256 + threadIdx.x;
    if (i < n) p[i] = 0.0f;
}

// ------------------------------------------------------------- scatter (edges)
// One wave (32 lanes) per edge; each lane atomically adds 4 features.
__global__ __launch_bounds__(256) void scatter_add(const float* __restrict__ H,
                                                   const int* __restrict__ ei,
                                                   float* __restrict__ Agg, int E) {
    int lane = threadIdx.x & 31;
    int e = blockIdx.x * 8 + (threadIdx.x >> 5);
    if (e >= E) return;
    int src = ei[e];       // edge_index[0][e]
    int dst = ei[E + e];   // edge_index[1][e]
    const float* hs = H + (long long)src * HIDF;
    float* ad = Agg + (long long)dst * HIDF;
#pragma unroll
    for (int k = 0; k < 4; ++k)
        atomicAdd(&ad[lane + 32 * k], hs[lane + 32 * k]);
}

// ----------------------------------------------------------------- WMMA GEMM
// Out[M,128] = (relu?)(Z[M,128] @ W[128,128] + bias).  One wave = one 16-row
// strip across all 8 column tiles.  M must be a multiple of 16 (50000 is).
//
// A 16x4 f32 layout: lanes 0-15 hold rows M=0..15; VGPR0 = K=0 (lo lanes) /
// K=2 (hi lanes), VGPR1 = K=1 / K=3.  B 4x16: cols across lanes, K striped
// the mirrored way.  C/D: VGPR r = row r (lo lanes) / row r+8 (hi lanes).
__global__ __launch_bounds__(128) void gemm_wmma_f32(const float* __restrict__ Z,
                                                     const float* __restrict__ W,
                                                     const float* __restrict__ bias,
                                                     float* __restrict__ Out,
                                                     int M, int do_relu) {
    int wave = threadIdx.x >> 5;
    int lane = threadIdx.x & 31;
    int strip = blockIdx.x * 4 + wave;
    if (strip * 16 >= M) return;              // whole-wave exit: EXEC stays all-1s
    int lane16 = lane & 15;
    int khalf  = lane >> 4;                   // 0 for lanes 0-15, 1 for 16-31
    int rowb   = strip * 16;

    v8f acc[8];
#pragma unroll
    for (int j = 0; j < 8; ++j) {
        float b = bias[j * 16 + lane16];
#pragma unroll
        for (int r = 0; r < 8; ++r) acc[j][r] = b;
    }

    const float* arow = Z + (long long)(rowb + lane16) * HIDF + 2 * khalf;

#pragma unroll 4
    for (int k0 = 0; k0 < HIDF; k0 += 4) {
        v2f a = *(const v2f*)(arow + k0);                  // K = k0+2*khalf, +1
        const float* wp = W + (k0 + 2 * khalf) * HIDF + lane16;
#pragma unroll
        for (int j = 0; j < 8; ++j) {
            v2f b;
            b.x = wp[j * 16];                              // W[k][n]
            b.y = wp[j * 16 + HIDF];                       // W[k+1][n]
            acc[j] = __builtin_amdgcn_wmma_f32_16x16x4_f32(
                false, a, false, b, (short)0, acc[j], false, false);
        }
    }

#pragma unroll
    for (int j = 0; j < 8; ++j) {
#pragma unroll
        for (int r = 0; r < 8; ++r) {
            float v = acc[j][r];
            if (do_relu) v = v > 0.0f ? v : 0.0f;
            Out[(long long)(rowb + r + 8 * khalf) * HIDF + j * 16 + lane16] = v;
        }
    }
}

// ------------------------------------------------------------ BN column stats
// stats[0:128] = sum over rows, stats[128:256] = sum of squares.
__global__ __launch_bounds__(128) void col_stats(const float* __restrict__ Y,
                                                 float* __restrict__ stats, int M) {
    int f = threadIdx.x;  // 128 threads = 128 features
    float s = 0.0f, s2 = 0.0f;
    for (int r = blockIdx.x; r < M; r += gridDim.x) {
        float v = Y[(long long)r * HIDF + f];
        s += v;
        s2 += v * v;
    }
    atomicAdd(&stats[f], s);
    atomicAdd(&stats[HIDF + f], s2);
}

// ------------------------------------------------------------- BN apply + ReLU
__global__ __launch_bounds__(256) void bn_relu(const float* __restrict__ Y,
                                               const float* __restrict__ stats,
                                               const float* __restrict__ g,
                                               const float* __restrict__ be,
                                               float* __restrict__ Out, int M) {
    int i = blockIdx.x * 256 + threadIdx.x;
    int total = M * HIDF;
    if (i >= total) return;
    int f = i & (HIDF - 1);
    float inv_m = 1.0f / (float)M;
    float mu  = stats[f] * inv_m;
    float var = stats[HIDF + f] * inv_m - mu * mu;     // biased, matches ref
    float rs  = rsqrtf(var + 1e-5f);
    float v = (Y[i] - mu) * rs * g[f] + be[f];
    Out[i] = v > 0.0f ? v : 0.0f;
}

// -------------------------------------------------------------------- launch
extern "C" void kernel_launch(void* const* d_in, const int* in_sizes, int n_in,
                              void* d_out, int out_size, void* d_ws, size_t ws_size,
                              hipStream_t stream) {
    const float* x    = (const float*)d_in[0];
    const int*   ei   = (const int*)d_in[1];
    const float* w1_0 = (const float*)d_in[2];
    const float* b1_0 = (const float*)d_in[3];
    const float* w2_0 = (const float*)d_in[4];
    const float* b2_0 = (const float*)d_in[5];
    const float* g_0  = (const float*)d_in[6];
    const float* be_0 = (const float*)d_in[7];
    const float* w1_1 = (const float*)d_in[8];
    const float* b1_1 = (const float*)d_in[9];
    const float* w2_1 = (const float*)d_in[10];
    const float* b2_1 = (const float*)d_in[11];
    const float* g_1  = (const float*)d_in[12];
    const float* be_1 = (const float*)d_in[13];

    const int M = in_sizes[0] / HIDF;   // 50000
    const int E = in_sizes[1] / 2;      // 800000
    float* out = (float*)d_out;

    size_t bufBytes = (size_t)M * HIDF * sizeof(float);
    char*  ws = (char*)d_ws;
    float* B0    = (float*)ws;                       // 25.6 MB
    float* B1    = (float*)(ws + bufBytes);          // 25.6 MB
    float* stats = (float*)(ws + 2 * bufBytes);      // 256 floats

    const int nelem    = M * HIDF;
    const int ewBlocks = (nelem + 255) / 256;
    const int scBlocks = (E + 7) / 8;
    const int gmBlocks = ((M / 16) + 3) / 4;         // 4 waves/block, 1 strip each
    const int stBlocks = 512;

    // ---------------- layer 0 :  h = x ----------------
    copy_f32<<<ewBlocks, 256, 0, stream>>>(x, B0, nelem);               // agg = h
    scatter_add<<<scBlocks, 256, 0, stream>>>(x, ei, B0, E);            // agg += Σ h[src]
    gemm_wmma_f32<<<gmBlocks, 128, 0, stream>>>(B0, w1_0, b1_0, B1, M, 1); // t = relu(z@w1+b1)
    gemm_wmma_f32<<<gmBlocks, 128, 0, stream>>>(B1, w2_0, b2_0, out, M, 0); // y = t@w2+b2
    zero_f32<<<1, 256, 0, stream>>>(stats, 2 * HIDF);
    col_stats<<<stBlocks, 128, 0, stream>>>(out, stats, M);
    bn_relu<<<ewBlocks, 256, 0, stream>>>(out, stats, g_0, be_0, B0, M);   // h1 -> B0

    // ---------------- layer 1 :  h = B0 ----------------
    copy_f32<<<ewBlocks, 256, 0, stream>>>(B0, B1, nelem);              // agg = h1
    scatter_add<<<scBlocks, 256, 0, stream>>>(B0, ei, B1, E);
    gemm_wmma_f32<<<gmBlocks, 128, 0, stream>>>(B1, w1_1, b1_1, out, M, 1); // t -> out (temp)
    gemm_wmma_f32<<<gmBlocks, 128, 0, stream>>>(out, w2_1, b2_1, B0, M, 0); // y -> B0
    zero_f32<<<1, 256, 0, stream>>>(stats, 2 * HIDF);
    col_stats<<<stBlocks, 128, 0, stream>>>(B0, stats, M);
    bn_relu<<<ewBlocks, 256, 0, stream>>>(B0, stats, g_1, be_1, out, M);   // final -> d_out
}